// SmileModel_54683523613254
// MI455X (gfx1250) — compile-verified
//
#include <hip/hip_runtime.h>
#include <hip/hip_bf16.h>

// ---------------------------------------------------------------------------
// SmileModel fwd + analytic forward-mode derivatives (d/dttm, d/dlogm, d2/dlogm2)
// One sample per lane, fp32, transcendental-heavy. Layer-2 (the only true 5x5
// dense matmul) is mapped to V_WMMA_F32_16X16X4_F32: A = W2 (loop invariant),
// B = 16 samples' activations per column group, cross-half marshalling via a
// single ds_swizzle SWAPX16 per register.
// ---------------------------------------------------------------------------

typedef float v2f __attribute__((ext_vector_type(2)));
typedef float v8f __attribute__((ext_vector_type(8)));

#define LOG2E 1.44269504088896340736f
#define LN2   0.69314718055994530942f

__device__ __forceinline__ float fexp(float x)  { return __builtin_amdgcn_exp2f(x * LOG2E); }
__device__ __forceinline__ float flog2(float x) { return __builtin_amdgcn_logf(x); }
__device__ __forceinline__ float frcp(float x)  { return __builtin_amdgcn_rcpf(x); }
__device__ __forceinline__ float fsqrt(float x) { return __builtin_amdgcn_sqrtf(x); }

__device__ __forceinline__ float ftanh(float x) {
#if defined(__has_builtin)
#if __has_builtin(__builtin_amdgcn_tanhf)
  return __builtin_amdgcn_tanhf(x);
#else
  float e = __builtin_amdgcn_exp2f(x * (2.0f * LOG2E));   // e^{2x}
  return 1.0f - 2.0f * frcp(e + 1.0f);
#endif
#else
  float e = __builtin_amdgcn_exp2f(x * (2.0f * LOG2E));
  return 1.0f - 2.0f * frcp(e + 1.0f);
#endif
}

// lane <-> lane^16 swap across the wave32 halves (ds_swizzle SWAPX16)
__device__ __forceinline__ float swap16(float x) {
  return __int_as_float(__builtin_amdgcn_ds_swizzle(__float_as_int(x), 0x401F));
}

// softplus value and sigmoid (= softplus') in one exp
__device__ __forceinline__ void softplus_sig(float z, float& sp, float& sg) {
  float en = __builtin_amdgcn_exp2f(-z * LOG2E);          // e^{-z}
  sg = frcp(1.0f + en);
  float a = (z > 0.0f) ? en : frcp(en);                   // e^{-|z|}
  float m = (z > 0.0f) ? z : 0.0f;
  sp = m + LN2 * flog2(1.0f + a);
}

#if defined(__has_builtin)
#if __has_builtin(__builtin_amdgcn_wmma_f32_16x16x4_f32)
#define HAVE_WMMA_F32X4 1
#endif
#endif

__global__ __launch_bounds__(256) void smile_fused_kernel(
    const float* __restrict__ ttm,   const float* __restrict__ logm,
    const float* __restrict__ wlg,   const float* __restrict__ blg,
    const float* __restrict__ wtt,   const float* __restrict__ btt,
    const float* __restrict__ wex,   const float* __restrict__ bex,
    const float* __restrict__ W1,    const float* __restrict__ b1,
    const float* __restrict__ W2,    const float* __restrict__ b2,
    const float* __restrict__ Wo,    const float* __restrict__ bo,
    float* __restrict__ out, int n)
{
  int gid = blockIdx.x * blockDim.x + threadIdx.x;
  int i   = (gid < n) ? gid : (n - 1);            // clamp: keep EXEC full for WMMA
  const float l = logm[i];
  const float t = ttm[i];

  // ---------------- front end: smile transform, forward-mode streams --------
  // u = sum_j term_logm_j * term_ttm_j * exp(wex_j); streams: value,d1,d2,dt
  float uv = 0.0f, u1 = 0.0f, u2 = 0.0f, ut = 0.0f;
#pragma unroll
  for (int j = 0; j < 5; ++j) {
    float wl = fexp(wlg[j]);
    float x  = blg[j] + l * wl;
    float t1 = ftanh(x + 0.5f);
    float t2 = ftanh(0.5f * x);
    float o1 = 1.0f - t1 * t1;
    float o2 = 1.0f - t2 * t2;
    float g   = x * t1 - t2 + 5.0e-4f;
    float gp  = t1 + x * o1 - 0.5f * o2;
    float gpp = 2.0f * o1 - 2.0f * x * t1 * o1 + 0.5f * t2 * o2;
    float s     = fsqrt(g);
    float inv2s = 0.5f * frcp(s);
    float sp_   = gp * inv2s;                               // ds/dx
    float spp   = inv2s * (gpp - gp * gp * 0.5f * frcp(g)); // d2s/dx2
    float sl1 = sp_ * wl;                                   // d/dlogm
    float sl2 = spp * wl * wl;                              // d2/dlogm2
    float wt = fexp(wtt[j]);
    float zt = btt[j] + t * wt;
    float q  = frcp(1.0f + fexp(-zt));                      // sigmoid
    float qt = q * (1.0f - q) * wt;                         // d/dttm
    float we = fexp(wex[j]);
    uv += s   * q  * we;
    u1 += sl1 * q  * we;
    u2 += sl2 * q  * we;
    ut += s   * qt * we;
  }

  // ---------------- layer 1 collapses to rank-1 (input = u + bex[j]) --------
  float h1[4][5];   // streams: [0]=value [1]=d/dlogm [2]=d2/dlogm2 [3]=d/dttm
#pragma unroll
  for (int r = 0; r < 5; ++r) {
    float rs = 0.0f, c1 = b1[r];
#pragma unroll
    for (int j = 0; j < 5; ++j) { float w = W1[r * 5 + j]; rs += w; c1 += w * bex[j]; }
    float z   = rs * uv + c1;
    float zd1 = rs * u1, zd2 = rs * u2, zdt = rs * ut;
    float sp, sg; softplus_sig(z, sp, sg);
    h1[0][r] = sp;
    h1[1][r] = sg * zd1;
    h1[2][r] = sg * (1.0f - sg) * zd1 * zd1 + sg * zd2;
    h1[3][r] = sg * zdt;
  }

  // ---------------- layer 2: the real 5x5 matmul, 4 streams -----------------
  float z2[4][5];
  const int lane = threadIdx.x & 31;
  const bool lo  = (lane & 16) == 0;

#if defined(HAVE_WMMA_F32X4)
  // A = W2 in the 16x4 f32 WMMA layout (row = lane&15; K pair by half-wave)
  {
    int row = lane & 15;
    int kb  = lo ? 0 : 2;
    int rr  = (row < 5) ? row : 4;                 // clamp, mask below
    float a0 = W2[rr * 5 + kb];
    float a1 = W2[rr * 5 + kb + 1];
    if (row >= 5) { a0 = 0.0f; a1 = 0.0f; }
    v2f A = { a0, a1 };

#pragma unroll
    for (int st = 0; st < 4; ++st) {
      float s0 = h1[st][0], s1 = h1[st][1], s2 = h1[st][2], s3 = h1[st][3], s4 = h1[st][4];
      float x0 = swap16(s0), x1 = swap16(s1), x2 = swap16(s2), x3 = swap16(s3), x4 = swap16(s4);
      // B (4x16): cols = samples; group0 = samples(lanes)0..15, group1 = 16..31
      v2f B0 = { lo ? s0 : x2, lo ? s1 : x3 };
      v2f B1 = { lo ? x0 : s2, lo ? x1 : s3 };
      // C carries bias (value stream only) + the K=4 leftover column of W2
      v8f C0 = {0.f,0.f,0.f,0.f,0.f,0.f,0.f,0.f};
      v8f C1 = {0.f,0.f,0.f,0.f,0.f,0.f,0.f,0.f};
#pragma unroll
      for (int r = 0; r < 5; ++r) {
        float base = (st == 0) ? b2[r] : 0.0f;
        float w24  = W2[r * 5 + 4];
        C0[r] = base + w24 * s4;    // group0 columns' h[4] already in lanes 0-15
        C1[r] = base + w24 * x4;    // group1 columns' h[4] swapped into lanes 0-15
      }
      v8f D0 = __builtin_amdgcn_wmma_f32_16x16x4_f32(false, A, false, B0, (short)0, C0, false, false);
      v8f D1 = __builtin_amdgcn_wmma_f32_16x16x4_f32(false, A, false, B1, (short)0, C1, false, false);
      // unpack D (row r, col = sample) back to owner lanes
#pragma unroll
      for (int r = 0; r < 5; ++r) {
        float hiv = swap16(D1[r]);
        z2[st][r] = lo ? D0[r] : hiv;
      }
    }
  }
#else
  // VALU fallback (kept so compilation never breaks if builtin is absent)
#pragma unroll
  for (int st = 0; st < 4; ++st)
#pragma unroll
    for (int r = 0; r < 5; ++r) {
      float acc = (st == 0) ? b2[r] : 0.0f;
#pragma unroll
      for (int j = 0; j < 5; ++j) acc += W2[r * 5 + j] * h1[st][j];
      z2[st][r] = acc;
    }
#endif

  // ---------------- layer-2 nonlinearity + output head ----------------------
  float ov = bo[0], o1s = 0.0f, o2s = 0.0f, ots = 0.0f;
#pragma unroll
  for (int r = 0; r < 5; ++r) {
    float z = z2[0][r];
    float sp, sg; softplus_sig(z, sp, sg);
    float zd1 = z2[1][r], zd2 = z2[2][r], zdt = z2[3][r];
    float hv = sp;
    float ha = sg * zd1;
    float hb = sg * (1.0f - sg) * zd1 * zd1 + sg * zd2;
    float ht = sg * zdt;
    float w  = Wo[r];
    ov  += w * hv;
    o1s += w * ha;
    o2s += w * hb;
    ots += w * ht;
  }

  // outputs concatenated: (output, grad_ttm1, grad_logm1, grad_logm2)
  out[i]         = ov;
  out[n + i]     = ots;
  out[2 * n + i] = o1s;
  out[3 * n + i] = o2s;
}

extern "C" void kernel_launch(void* const* d_in, const int* in_sizes, int n_in,
                              void* d_out, int out_size, void* d_ws, size_t ws_size,
                              hipStream_t stream) {
  (void)n_in; (void)out_size; (void)d_ws; (void)ws_size;
  int n = in_sizes[0];
  dim3 block(256);
  dim3 grid((unsigned)((n + 255) / 256));
  smile_fused_kernel<<<grid, block, 0, stream>>>(
      (const float*)d_in[0],  (const float*)d_in[1],
      (const float*)d_in[2],  (const float*)d_in[3],
      (const float*)d_in[4],  (const float*)d_in[5],
      (const float*)d_in[6],  (const float*)d_in[7],
      (const float*)d_in[8],  (const float*)d_in[9],
      (const float*)d_in[10], (const float*)d_in[11],
      (const float*)d_in[12], (const float*)d_in[13],
      (float*)d_out, n);
}